// MotionTracking_49555332661374
// MI455X (gfx1250) — compile-verified
//
#include <hip/hip_runtime.h>
#include <cstdint>
#include <cstddef>

#define NTRACKS 64
#define WIN 31
#define NP (WIN * WIN)   // 961 samples per window
#define TILE 34          // staged tile edge (33 needed + 1 ulp-safety)
#define TSTRIDE 35       // LDS row stride (pad against bank conflicts)
#define IMGH 512
#define IMGW 512
#define BLK 256

typedef __attribute__((address_space(1))) int* as1_int_p;  // global
typedef __attribute__((address_space(3))) int* as3_int_p;  // LDS

__device__ __forceinline__ int clampi(int v, int lo, int hi) {
    return v < lo ? lo : (v > hi ? hi : v);
}

// MotionTracking: bilinear window sampling (the only live output of the reference).
// Strategy: per (batch, track) block, stage the 34x34 border-clamped bounding box of
// the sample grid for both frames into LDS with gfx1250 async global->LDS loads,
// then compute 2x961 bilinear taps from LDS and store coalesced.
__global__ void __launch_bounds__(BLK)
MotionTracking_49555332661374_kernel(const float* __restrict__ track_locs,
                                     const float* __restrict__ imgs,
                                     float* __restrict__ out)
{
    __shared__ float tile[2][TILE][TSTRIDE];

    const int bt  = blockIdx.x;
    const int b   = bt >> 6;      // / NTRACKS
    const int t   = bt & 63;      // % NTRACKS
    const int tid = threadIdx.x;

    // track_locs: (64, NTRACKS*2) -> reshape (B, nt, 2): [b][t][0]=x, [b][t][1]=y
    const float tlx = track_locs[b * (NTRACKS * 2) + 2 * t + 0];
    const float tly = track_locs[b * (NTRACKS * 2) + 2 * t + 1];

    // Reference math: x(col) = (lin[col] + tlx*512/31) * 15.5, lin[col] = -1 + col*(2/30)
    const float offx = tlx * (512.0f / 31.0f);
    const float offy = tly * (512.0f / 31.0f);
    const float xs0  = (-1.0f + offx) * (0.5f * (float)WIN);  // sample x at col=0 (minimum)
    const float ys0  = (-1.0f + offy) * (0.5f * (float)WIN);

    const int tile_x0 = (int)floorf(xs0);
    const int tile_y0 = (int)floorf(ys0);

    // ---- Stage both frames' clamped tiles into LDS (async global -> LDS) ----
    const float* img_b = imgs + (size_t)(b * 2) * (IMGH * IMGW);
    for (int f = 0; f < 2; ++f) {
        const float* img = img_b + (size_t)f * (IMGH * IMGW);
        for (int i = tid; i < TILE * TILE; i += BLK) {
            const int r  = i / TILE;
            const int c  = i - r * TILE;
            const int gy = clampi(tile_y0 + r, 0, IMGH - 1);
            const int gx = clampi(tile_x0 + c, 0, IMGW - 1);
            const float* gp = img + (size_t)gy * IMGW + gx;
            float*       lp = &tile[f][r][c];
#if __has_builtin(__builtin_amdgcn_global_load_async_to_lds_b32)
            __builtin_amdgcn_global_load_async_to_lds_b32(
                (as1_int_p)(void*)const_cast<float*>(gp),
                (as3_int_p)(void*)lp,
                /*imm offset*/ 0, /*cpol*/ 0);
#else
            *lp = *gp;   // fallback: plain load + ds_store
#endif
        }
    }
#if __has_builtin(__builtin_amdgcn_global_load_async_to_lds_b32)
#if __has_builtin(__builtin_amdgcn_s_wait_asynccnt)
    __builtin_amdgcn_s_wait_asynccnt(0);
#else
    asm volatile("s_wait_asynccnt 0" ::: "memory");
#endif
#endif
    __syncthreads();

    // ---- Bilinear taps from LDS, exact reference weight formulation ----
    // out: (B, 2, nt, 31, 31, 1)
    float* out0 = out + ((size_t)(b * 2 + 0) * NTRACKS + t) * NP;
    float* out1 = out + ((size_t)(b * 2 + 1) * NTRACKS + t) * NP;

    for (int p = tid; p < NP; p += BLK) {
        const int row = p / WIN;
        const int col = p - row * WIN;

        const float linx = -1.0f + (float)col * (2.0f / 30.0f);
        const float liny = -1.0f + (float)row * (2.0f / 30.0f);
        const float x = (linx + offx) * (0.5f * (float)WIN);
        const float y = (liny + offy) * (0.5f * (float)WIN);

        const float fx = floorf(x);
        const float fy = floorf(y);
        const float x0f = fminf(fmaxf(fx,        0.0f), (float)(IMGW - 1));
        const float x1f = fminf(fmaxf(fx + 1.0f, 0.0f), (float)(IMGW - 1));
        const float y0f = fminf(fmaxf(fy,        0.0f), (float)(IMGH - 1));
        const float y1f = fminf(fmaxf(fy + 1.0f, 0.0f), (float)(IMGH - 1));

        const float wx1 = x1f - x;   // weight for x0 taps
        const float wx0 = x - x0f;   // weight for x1 taps
        const float wy1 = y1f - y;   // weight for y0 taps
        const float wy0 = y - y0f;   // weight for y1 taps

        // LDS tile column c holds img[clamp(tile_x0 + c)] == the clipped tap index
        const int c0 = clampi((int)fx - tile_x0, 0, TILE - 1);
        const int c1 = clampi(c0 + 1,            0, TILE - 1);
        const int r0 = clampi((int)fy - tile_y0, 0, TILE - 1);
        const int r1 = clampi(r0 + 1,            0, TILE - 1);

        // frame 0
        {
            const float Ia = tile[0][r0][c0];
            const float Ib = tile[0][r0][c1];
            const float Ic = tile[0][r1][c0];
            const float Id = tile[0][r1][c1];
            out0[p] = wy1 * (wx1 * Ia + wx0 * Ib) + wy0 * (wx1 * Ic + wx0 * Id);
        }
        // frame 1
        {
            const float Ia = tile[1][r0][c0];
            const float Ib = tile[1][r0][c1];
            const float Ic = tile[1][r1][c0];
            const float Id = tile[1][r1][c1];
            out1[p] = wy1 * (wx1 * Ia + wx0 * Ib) + wy0 * (wx1 * Ic + wx0 * Id);
        }
    }
}

extern "C" void kernel_launch(void* const* d_in, const int* in_sizes, int n_in,
                              void* d_out, int out_size, void* d_ws, size_t ws_size,
                              hipStream_t stream) {
    (void)in_sizes; (void)n_in; (void)out_size; (void)d_ws; (void)ws_size;
    const float* track_locs = (const float*)d_in[0];  // (64, 128) f32
    const float* imgs       = (const float*)d_in[1];  // (64, 2, 512, 512, 1) f32
    float*       out        = (float*)d_out;          // (64, 2, 64, 31, 31, 1) f32

    const int nblocks = 64 * NTRACKS;                 // one block per (batch, track)
    MotionTracking_49555332661374_kernel<<<nblocks, BLK, 0, stream>>>(track_locs, imgs, out);
}